// ETR_loss_trace_69672959476240
// MI455X (gfx1250) — compile-verified
//
#include <hip/hip_runtime.h>
#include <math.h>

// ---------------------------------------------------------------------------
// ETR loss for MI455X (gfx1250, wave32, WMMA).
//
// Inputs (setup_inputs order):
//   0 A_t [n,n] f32   1 C [n,k] f32   2 B_t [n,n] f32   3 X [n,d] f32
//   4 rate_a (1,)     5 rate_b (1,)   6 x2 (unused)
//   7 IsumC [k,n]     8 IsumCDC [k,k] 9 Amask [n,n]
// Output: 5 floats (total, total1, total2, cons, fea_loss)
//
// Workspace requirement: ~10.6 MB (floats, see layout below).
// ---------------------------------------------------------------------------

#define N_DIM 8192
#define K_DIM 32
#define D_DIM 256
#define JSPLIT 4
#define JCHUNK (N_DIM / JSPLIT)      // 2048 columns per wave
#define NSTRIP (N_DIM / 16)          // 512 row strips
#define NWAVE1 (NSTRIP * JSPLIT)     // 2048 waves in kernel 1
#define EPS_LOG 1e-20f

typedef __attribute__((ext_vector_type(2))) float v2f;
typedef __attribute__((ext_vector_type(8))) float v8f;

// ---------------------------------------------------------------------------
// Kernel 1: fused pass over A_t, B_t, Amask.
// One wave per (16-row strip, j-chunk). Computes:
//   partial MC_A = (A_t*mask)@C and MC_B = (B_t*mask)@C via V_WMMA_F32_16X16X4_F32
//   per-wave partials of sum(A*mask), sum(B*mask), sum(((A*B-B)*mask)^2)
// A-operand layout (ISA 7.12.2, 32-bit A 16x4): lane L<16 holds row L, K=0,1;
// lane L+16 holds row L, K=2,3  -> one contiguous b64 load per matrix.
// B-operand (4x16): lane L holds N=L&15 (C col +16 for c1), K-pair (0,1) or (2,3).
// ---------------------------------------------------------------------------
__global__ __launch_bounds__(32) void
k1_fused(const float* __restrict__ A, const float* __restrict__ B,
         const float* __restrict__ Mk, const float* __restrict__ C,
         float* __restrict__ partA, float* __restrict__ partB,
         float* __restrict__ scal /* [NWAVE1*3] */) {
  const int strip = blockIdx.x;            // 0..511
  const int p     = blockIdx.y;            // 0..JSPLIT-1
  const int lane  = threadIdx.x;           // 0..31
  const int rowL  = lane & 15;
  const int koff  = (lane >> 4) << 1;      // 0 or 2
  const size_t rbase = (size_t)(strip * 16 + rowL) * N_DIM;

  v8f accA0 = {}, accA1 = {}, accB0 = {}, accB1 = {};
  float sA = 0.f, sB = 0.f, cs = 0.f;

  const int j0 = p * JCHUNK;
  for (int j = j0; j < j0 + JCHUNK; j += 4) {
    const int jj = j + koff;
    float2 a = *(const float2*)(A  + rbase + jj);
    float2 b = *(const float2*)(B  + rbase + jj);
    float2 m = *(const float2*)(Mk + rbase + jj);

    float ax = a.x * m.x, ay = a.y * m.y;
    float bx = b.x * m.x, by = b.y * m.y;
    sA += ax + ay;
    sB += bx + by;
    float e0 = (a.x * b.x - b.x) * m.x;
    float e1 = (a.y * b.y - b.y) * m.y;
    cs += e0 * e0 + e1 * e1;

    v2f av; av[0] = ax; av[1] = ay;
    v2f bv; bv[0] = bx; bv[1] = by;
    // C rows jj, jj+1; N = rowL (+16 for c1)
    v2f c0, c1;
    c0[0] = C[(size_t)jj * K_DIM + rowL];
    c0[1] = C[(size_t)(jj + 1) * K_DIM + rowL];
    c1[0] = C[(size_t)jj * K_DIM + 16 + rowL];
    c1[1] = C[(size_t)(jj + 1) * K_DIM + 16 + rowL];

#if defined(__HIP_DEVICE_COMPILE__)
    accA0 = __builtin_amdgcn_wmma_f32_16x16x4_f32(false, av, false, c0, (short)0, accA0, false, false);
    accA1 = __builtin_amdgcn_wmma_f32_16x16x4_f32(false, av, false, c1, (short)0, accA1, false, false);
    accB0 = __builtin_amdgcn_wmma_f32_16x16x4_f32(false, bv, false, c0, (short)0, accB0, false, false);
    accB1 = __builtin_amdgcn_wmma_f32_16x16x4_f32(false, bv, false, c1, (short)0, accB1, false, false);
#endif
  }

  // D layout: VGPR r -> lanes 0-15: (M=r, N=lane); lanes 16-31: (M=r+8, N=lane-16)
  const int outRow0 = strip * 16 + ((lane >> 4) << 3);
  const int nIdx = lane & 15;
  float* pa = partA + (size_t)p * N_DIM * K_DIM;
  float* pb = partB + (size_t)p * N_DIM * K_DIM;
#pragma unroll
  for (int r = 0; r < 8; ++r) {
    size_t ro = (size_t)(outRow0 + r) * K_DIM;
    pa[ro + nIdx]      = accA0[r];
    pa[ro + 16 + nIdx] = accA1[r];
    pb[ro + nIdx]      = accB0[r];
    pb[ro + 16 + nIdx] = accB1[r];
  }

  // wave-reduce scalars (fixed order -> deterministic per wave)
  for (int off = 16; off > 0; off >>= 1) {
    sA += __shfl_xor(sA, off, 32);
    sB += __shfl_xor(sB, off, 32);
    cs += __shfl_xor(cs, off, 32);
  }
  if (lane == 0) {
    int wid = p * NSTRIP + strip;
    scal[wid * 3 + 0] = sA;
    scal[wid * 3 + 1] = sB;
    scal[wid * 3 + 2] = cs;
  }
}

// Sum the JSPLIT partials into final MC (deterministic fixed order).
__global__ __launch_bounds__(256) void
k1b_sumpart(const float* __restrict__ part, float* __restrict__ mc) {
  size_t idx = (size_t)blockIdx.x * 256 + threadIdx.x;
  const size_t NK = (size_t)N_DIM * K_DIM;
  float s = part[idx] + part[idx + NK] + part[idx + 2 * NK] + part[idx + 3 * NK];
  mc[idx] = s;
}

// ---------------------------------------------------------------------------
// Kernel 2: per-column stats. Csum[i] = sum_r C[r,i];
// diagR_{A,B}[i] = sum_r C[r,i]*MC_{A,B}[r,i]   (diagonal of C.T@MC)
// ---------------------------------------------------------------------------
__global__ __launch_bounds__(256) void
k2_colstats(const float* __restrict__ C, const float* __restrict__ mcA,
            const float* __restrict__ mcB, float* __restrict__ Csum,
            float* __restrict__ dRA, float* __restrict__ dRB) {
  const int i = blockIdx.x, t = threadIdx.x;
  float s0 = 0.f, s1 = 0.f, s2 = 0.f;
  for (int r = t; r < N_DIM; r += 256) {
    size_t o = (size_t)r * K_DIM + i;
    float c = C[o];
    s0 += c;
    s1 += c * mcA[o];
    s2 += c * mcB[o];
  }
  __shared__ float l0[256], l1[256], l2[256];
  l0[t] = s0; l1[t] = s1; l2[t] = s2;
  __syncthreads();
  for (int s = 128; s > 0; s >>= 1) {
    if (t < s) { l0[t] += l0[t + s]; l1[t] += l1[t + s]; l2[t] += l2[t + s]; }
    __syncthreads();
  }
  if (t == 0) { Csum[i] = l0[0]; dRA[i] = l1[0]; dRB[i] = l2[0]; }
}

// ---------------------------------------------------------------------------
// Kernel 3: P[a,i] = sum_r IsumC[a,r]*MC[r,i]   ([k,k] = [k,n]@[n,k])
// ---------------------------------------------------------------------------
__global__ __launch_bounds__(256) void
k3_P(const float* __restrict__ IsumC, const float* __restrict__ mc,
     float* __restrict__ P) {
  const int a = blockIdx.x, i = blockIdx.y, t = threadIdx.x;
  float s = 0.f;
  for (int r = t; r < N_DIM; r += 256)
    s += IsumC[(size_t)a * N_DIM + r] * mc[(size_t)r * K_DIM + i];
  __shared__ float l[256];
  l[t] = s;
  __syncthreads();
  for (int st = 128; st > 0; st >>= 1) {
    if (t < st) l[t] += l[t + st];
    __syncthreads();
  }
  if (t == 0) P[a * K_DIM + i] = l[0];
}

// ---------------------------------------------------------------------------
// Kernel 4: CtX[i,j] = sum_r C[r,i]*X[r,j]; G[i,j] = sum_r C[r,i]^2*X[r,j]
// grid (32, 4); block 256 = 64 j-columns x 4 r-groups.
// ---------------------------------------------------------------------------
__global__ __launch_bounds__(256) void
k4_ctx(const float* __restrict__ C, const float* __restrict__ X,
       float* __restrict__ CtX, float* __restrict__ G) {
  const int i = blockIdx.x;
  const int j = blockIdx.y * 64 + (threadIdx.x & 63);
  const int rg = threadIdx.x >> 6;
  float g0 = 0.f, g1 = 0.f;
  for (int r = rg; r < N_DIM; r += 4) {
    float c = C[(size_t)r * K_DIM + i];
    float x = X[(size_t)r * D_DIM + j];
    g0 += c * x;
    g1 += c * c * x;
  }
  __shared__ float s0[256], s1[256];
  s0[threadIdx.x] = g0; s1[threadIdx.x] = g1;
  __syncthreads();
  if (rg == 0) {
    int jl = threadIdx.x;
    g0 = s0[jl] + s0[jl + 64] + s0[jl + 128] + s0[jl + 192];
    g1 = s1[jl] + s1[jl + 64] + s1[jl + 128] + s1[jl + 192];
    CtX[i * D_DIM + j] = g0;
    G[i * D_DIM + j] = g1;
  }
}

// Kernel 5: xsq[r] = sum_j X[r,j]^2  (one wave per row)
__global__ __launch_bounds__(256) void
k5_xsq(const float* __restrict__ X, float* __restrict__ xsq) {
  const int wid = threadIdx.x >> 5, lane = threadIdx.x & 31;
  const int r = blockIdx.x * 8 + wid;
  float s = 0.f;
#pragma unroll
  for (int t = 0; t < 8; ++t) {
    float x = X[(size_t)r * D_DIM + lane + t * 32];
    s += x * x;
  }
  for (int off = 16; off > 0; off >>= 1) s += __shfl_xor(s, off, 32);
  if (lane == 0) xsq[r] = s;
}

// Kernel 6: t3[i] = sum_r C[r,i]^2 * xsq[r]  (== row-sum of Csq.T@(X*X))
__global__ __launch_bounds__(256) void
k6_t3(const float* __restrict__ C, const float* __restrict__ xsq,
      float* __restrict__ t3) {
  const int i = blockIdx.x, t = threadIdx.x;
  float s = 0.f;
  for (int r = t; r < N_DIM; r += 256) {
    float c = C[(size_t)r * K_DIM + i];
    s += c * c * xsq[r];
  }
  __shared__ float l[256];
  l[t] = s;
  __syncthreads();
  for (int st = 128; st > 0; st >>= 1) {
    if (t < st) l[t] += l[t + st];
    __syncthreads();
  }
  if (t == 0) t3[i] = l[0];
}

// ---------------------------------------------------------------------------
// Kernel 7: DBI finalize (single block, 512 threads, 16 waves).
// ---------------------------------------------------------------------------
__global__ __launch_bounds__(512) void
k7_dbi(const float* __restrict__ CtX, const float* __restrict__ G,
       const float* __restrict__ Csum, const float* __restrict__ t3,
       float* __restrict__ fea) {
  __shared__ float sCen[K_DIM * D_DIM];   // 32 KB
  __shared__ float sCsum[K_DIM], sCn[K_DIM], sT2[K_DIM], sS[K_DIM];
  __shared__ float sR[K_DIM * K_DIM], sMax[K_DIM];
  const int t = threadIdx.x;
  if (t < K_DIM) sCsum[t] = Csum[t];
  __syncthreads();
  for (int idx = t; idx < K_DIM * D_DIM; idx += 512)
    sCen[idx] = CtX[idx] / sCsum[idx >> 8];
  __syncthreads();

  const int wid = t >> 5, lane = t & 31;
  for (int i = wid; i < K_DIM; i += 16) {
    float cn = 0.f, t2 = 0.f;
#pragma unroll
    for (int tt = 0; tt < 8; ++tt) {
      float ce = sCen[i * D_DIM + lane + tt * 32];
      float g  = G[i * D_DIM + lane + tt * 32];
      cn += ce * ce;
      t2 += ce * g;
    }
    for (int off = 16; off > 0; off >>= 1) {
      cn += __shfl_xor(cn, off, 32);
      t2 += __shfl_xor(t2, off, 32);
    }
    if (lane == 0) { sCn[i] = cn; sT2[i] = t2; }
  }
  __syncthreads();
  if (t < K_DIM) {
    float cs = sCsum[t];
    // S = sqrt((t1 + t2 + t3)/Csum), t1 = Csum*cn, t2 = -2*sum(Cen*G)
    sS[t] = sqrtf((cs * sCn[t] - 2.0f * sT2[t] + t3[t]) / cs);
  }
  __syncthreads();
  for (int pr = t; pr < K_DIM * K_DIM; pr += 512) {
    int i = pr >> 5, l = pr & 31;
    float dot = 0.f;
    for (int j = 0; j < D_DIM; ++j)
      dot += sCen[i * D_DIM + j] * sCen[l * D_DIM + j];
    float msq = sCn[i] + sCn[l] - 2.0f * dot;
    sR[pr] = (i == l) ? 0.0f : (sS[i] + sS[l]) / sqrtf(msq);
  }
  __syncthreads();
  if (t < K_DIM) {
    float m = sR[t];
    for (int i = 1; i < K_DIM; ++i) m = fmaxf(m, sR[i * K_DIM + t]);
    sMax[t] = m;
  }
  __syncthreads();
  if (t == 0) {
    float s = 0.f;
    for (int l = 0; l < K_DIM; ++l) s += sMax[l];
    *fea = s / (float)K_DIM;
  }
}

// ---------------------------------------------------------------------------
// Kernel 8: reduce per-wave scalars + assemble the 5 outputs.
// total_t = sum_i diagR[i]*deno * log2(deno * sum_a IsumCDC[i,a]*P[a,i] + EPS)
// ---------------------------------------------------------------------------
__global__ __launch_bounds__(256) void
k8_final(const float* __restrict__ scal, const float* __restrict__ dRA,
         const float* __restrict__ dRB, const float* __restrict__ PA,
         const float* __restrict__ PB, const float* __restrict__ IsumCDC,
         const float* __restrict__ fea, const float* __restrict__ rate_a,
         const float* __restrict__ rate_b, float* __restrict__ out) {
  const int t = threadIdx.x;
  float a = 0.f, b = 0.f, c = 0.f;
  for (int w = t; w < NWAVE1; w += 256) {
    a += scal[w * 3 + 0];
    b += scal[w * 3 + 1];
    c += scal[w * 3 + 2];
  }
  __shared__ float rA[256], rB[256], rC[256];
  rA[t] = a; rB[t] = b; rC[t] = c;
  __syncthreads();
  for (int s = 128; s > 0; s >>= 1) {
    if (t < s) { rA[t] += rA[t + s]; rB[t] += rB[t + s]; rC[t] += rC[t + s]; }
    __syncthreads();
  }
  __shared__ float sT1[32], sT2w[32];
  const float sumA = rA[0], sumB = rB[0], consSq = rC[0];
  __syncthreads();
  if (t < K_DIM) {
    float denoA = 1.0f / sumA, denoB = 1.0f / sumB;
    float eA = 0.f, eB = 0.f;
    for (int aa = 0; aa < K_DIM; ++aa) {
      float w = IsumCDC[t * K_DIM + aa];
      eA += w * PA[aa * K_DIM + t];
      eB += w * PB[aa * K_DIM + t];
    }
    sT1[t]  = dRA[t] * denoA * log2f(eA * denoA + EPS_LOG);
    sT2w[t] = dRB[t] * denoB * log2f(eB * denoB + EPS_LOG);
  }
  __syncthreads();
  if (t == 0) {
    float t1 = 0.f, t2 = 0.f;
    for (int i = 0; i < K_DIM; ++i) { t1 += sT1[i]; t2 += sT2w[i]; }
    float cons = sqrtf(consSq);
    float f = *fea;
    out[0] = t1 + t2 + (*rate_b) * f + (*rate_a) * cons;
    out[1] = t1;
    out[2] = t2;
    out[3] = cons;
    out[4] = f;
  }
}

// ---------------------------------------------------------------------------
extern "C" void kernel_launch(void* const* d_in, const int* in_sizes, int n_in,
                              void* d_out, int out_size, void* d_ws, size_t ws_size,
                              hipStream_t stream) {
  const float* A_t     = (const float*)d_in[0];
  const float* C       = (const float*)d_in[1];
  const float* B_t     = (const float*)d_in[2];
  const float* X       = (const float*)d_in[3];
  const float* rate_a  = (const float*)d_in[4];
  const float* rate_b  = (const float*)d_in[5];
  /* d_in[6] = x2, unused */
  const float* IsumC   = (const float*)d_in[7];
  const float* IsumCDC = (const float*)d_in[8];
  const float* Amask   = (const float*)d_in[9];
  float* out = (float*)d_out;

  // workspace layout (floats)
  const size_t NK = (size_t)N_DIM * K_DIM;   // 262144
  float* ws = (float*)d_ws;
  float* partA = ws;                         // 4*NK
  float* partB = partA + 4 * NK;             // 4*NK
  float* mcA   = partB + 4 * NK;             // NK
  float* mcB   = mcA + NK;                   // NK
  float* scal  = mcB + NK;                   // NWAVE1*3 = 6144
  float* Csum  = scal + (size_t)NWAVE1 * 3;  // 32
  float* dRA   = Csum + K_DIM;               // 32
  float* dRB   = dRA + K_DIM;                // 32
  float* PA    = dRB + K_DIM;                // 1024
  float* PB    = PA + K_DIM * K_DIM;         // 1024
  float* CtX   = PB + K_DIM * K_DIM;         // 8192
  float* G     = CtX + K_DIM * D_DIM;        // 8192
  float* xsq   = G + K_DIM * D_DIM;          // 8192
  float* t3    = xsq + N_DIM;                // 32
  float* fea   = t3 + K_DIM;                 // 1
  (void)in_sizes; (void)n_in; (void)out_size; (void)ws_size;

  // Phase 1: one fused pass over the three 256MB matrices (WMMA fp32 GEMM)
  k1_fused<<<dim3(NSTRIP, JSPLIT), 32, 0, stream>>>(A_t, B_t, Amask, C,
                                                    partA, partB, scal);
  k1b_sumpart<<<dim3((unsigned)(NK / 256)), 256, 0, stream>>>(partA, mcA);
  k1b_sumpart<<<dim3((unsigned)(NK / 256)), 256, 0, stream>>>(partB, mcB);

  // Phase 2: small L2-resident reductions
  k2_colstats<<<K_DIM, 256, 0, stream>>>(C, mcA, mcB, Csum, dRA, dRB);
  k3_P<<<dim3(K_DIM, K_DIM), 256, 0, stream>>>(IsumC, mcA, PA);
  k3_P<<<dim3(K_DIM, K_DIM), 256, 0, stream>>>(IsumC, mcB, PB);
  k4_ctx<<<dim3(K_DIM, D_DIM / 64), 256, 0, stream>>>(C, X, CtX, G);
  k5_xsq<<<N_DIM / 8, 256, 0, stream>>>(X, xsq);
  k6_t3<<<K_DIM, 256, 0, stream>>>(C, xsq, t3);
  k7_dbi<<<1, 512, 0, stream>>>(CtX, G, Csum, t3, fea);
  k8_final<<<1, 256, 0, stream>>>(scal, dRA, dRB, PA, PB, IsumCDC, fea,
                                  rate_a, rate_b, out);
}